// Transformer_69956427317618
// MI455X (gfx1250) — compile-verified
//
#include <hip/hip_runtime.h>
#include <hip/hip_bf16.h>
#include <math.h>

typedef __attribute__((ext_vector_type(2))) float v2f;
typedef __attribute__((ext_vector_type(8))) float v8f;

// ---------------- input indices (X,E,y,node_mask, then params in sorted-key order) --------------
enum {
  IN_X = 0, IN_E, IN_Y, IN_MASK,
  P_e_add_b, P_e_add_w, P_e_mul_b, P_e_mul_w,
  P_e_out_b, P_e_out_w, P_e_y_b, P_e_y_w,
  P_ffE1_b, P_ffE1_w, P_ffE2_b, P_ffE2_w,
  P_ffX1_b, P_ffX1_w, P_ffX2_b, P_ffX2_w,
  P_ffy1_b, P_ffy1_w, P_ffy2_b, P_ffy2_w,
  P_k_b, P_k_w,
  P_lnE1_b, P_lnE1_g, P_lnE2_b, P_lnE2_g,
  P_lnX1_b, P_lnX1_g, P_lnX2_b, P_lnX2_g,
  P_lny1_b, P_lny1_g, P_lny2_b, P_lny2_g,
  P_q_b, P_q_w, P_v_b, P_v_w,
  P_x_out_b, P_x_out_w, P_x_y_b, P_x_y_w,
  P_y_e_add_b, P_y_e_add_w, P_y_e_mul_b, P_y_e_mul_w,
  P_y_out1_b, P_y_out1_w, P_y_out2_b, P_y_out2_w,
  P_y_x_add_b, P_y_x_add_w, P_y_x_mul_b, P_y_x_mul_w,
  P_y_y_b, P_y_y_w
};

#define BSZ 4
#define NN 256
#define DXX 256
#define DEE 64
#define DYY 64

// ---------------- workspace layout (floats) ----------------
#define OFF_Q    ((size_t)0)
#define OFF_K    ((size_t)262144)
#define OFF_V    ((size_t)524288)
#define OFF_XA   ((size_t)786432)
#define OFF_XR   ((size_t)1048576)
#define OFF_XLN  ((size_t)1310720)
#define OFF_XH   ((size_t)1572864)
#define OFF_XR2  ((size_t)2621440)
#define OFF_YE1  ((size_t)2883584)
#define OFF_YE2  ((size_t)2884608)
#define OFF_YX1  ((size_t)2885632)
#define OFF_YX2  ((size_t)2886656)
#define OFF_ZX   ((size_t)2887680)
#define OFF_PM   ((size_t)2891776)
#define OFF_PS   ((size_t)2957312)
#define OFF_PMN  ((size_t)3022848)
#define OFF_PMX  ((size_t)3088384)
#define OFF_ZE   ((size_t)3153920)
#define OFF_ER   ((size_t)4194304)    // 16,777,216 floats; er2 aliases this
#define OFF_ELN  ((size_t)20971520)   // 16,777,216 floats
#define OFF_YBIG ((size_t)37748736)   // 67,108,864 floats; ffE hidden aliases this

#define OUT_XO   ((size_t)0)
#define OUT_EO   ((size_t)262144)
#define OUT_YO   ((size_t)17039360)

// ---------------- fp32 WMMA strip: C(16x64) = A(16xK) * W(64xK)^T -------------
// A fragment: lane L holds row (L&15), K-pair base ((L>>4)*2)
// B fragment: lane L holds col (L&15), same K-pair mapping
// C fragment: acc[v] at row v + 8*(L>>4), col (L&15)
// A fragment is loaded ONCE per K-step and reused across 4 column tiles.
__device__ __forceinline__ void wmma_f32_strip4(const float* __restrict__ Arow, int lda,
                                                const float* __restrict__ Wrow, int ldw,
                                                int K, v8f acc[4]) {
  const int lane = threadIdx.x & 31;
  const int r = lane & 15;
  const int kp = (lane >> 4) << 1;
  const float* a = Arow + (size_t)r * lda + kp;
  for (int k = 0; k < K; k += 4) {
    v2f av; av.x = a[k]; av.y = a[k + 1];
#pragma unroll
    for (int nt = 0; nt < 4; ++nt) {
      const float* w = Wrow + (size_t)(nt * 16 + r) * ldw + kp;
      v2f bv; bv.x = w[k]; bv.y = w[k + 1];
      acc[nt] = __builtin_amdgcn_wmma_f32_16x16x4_f32(false, av, false, bv,
                                                      (short)0, acc[nt], false, false);
    }
  }
}

// ---------------- generic GEMM: C = epi(A @ W^T + bias) ----------------
// one wave per 16x64 strip; grid = (N/64, M/16)
__global__ void k_gemm(const float* __restrict__ A, const float* __restrict__ W,
                       const float* __restrict__ bias, const float* __restrict__ resid,
                       const float* __restrict__ rowmask, float* __restrict__ C,
                       int N, int K, int relu) {
  const int lane = threadIdx.x;
  const int row0 = blockIdx.y << 4;
  const int col0 = blockIdx.x << 6;
  v8f acc[4] = {};
  wmma_f32_strip4(A + (size_t)row0 * K, K, W + (size_t)col0 * K, K, K, acc);
  const int n = lane & 15;
  const int mbase = (lane >> 4) << 3;
#pragma unroll
  for (int nt = 0; nt < 4; ++nt) {
    const int c = col0 + nt * 16 + n;
    const float bc = bias ? bias[c] : 0.f;
#pragma unroll
    for (int v = 0; v < 8; ++v) {
      const int row = row0 + mbase + v;
      float o = acc[nt][v] + bc;
      if (relu) o = fmaxf(o, 0.f);
      if (rowmask) o *= rowmask[row];
      const size_t idx = (size_t)row * N + c;
      if (resid) o += resid[idx];
      C[idx] = o;
    }
  }
}

// ------- fused E1/E2 GEMMs + QK FiLM: Y = E1 + E2*Yqk + Yqk (rows = b*n*n flat) -------
// one wave per 16x64 strip; A fragment shared across BOTH weight matrices (8 accumulators)
__global__ void k_e12(const float* __restrict__ E, const float* __restrict__ Wadd,
                      const float* __restrict__ badd, const float* __restrict__ Wmul,
                      const float* __restrict__ bmul, const float* __restrict__ Q,
                      const float* __restrict__ Km, const float* __restrict__ mask,
                      float* __restrict__ Ybig) {
  const int lane = threadIdx.x;
  const int row0 = blockIdx.y << 4;
  const int col0 = blockIdx.x << 6;
  const int r = lane & 15;
  const int kp = (lane >> 4) << 1;
  const float* a = E + (size_t)(row0 + r) * DEE + kp;
  v8f a1[4] = {}, a2[4] = {};
  for (int k = 0; k < DEE; k += 4) {
    v2f av; av.x = a[k]; av.y = a[k + 1];
#pragma unroll
    for (int nt = 0; nt < 4; ++nt) {
      const float* w1 = Wadd + (size_t)(col0 + nt * 16 + r) * DEE + kp;
      const float* w2 = Wmul + (size_t)(col0 + nt * 16 + r) * DEE + kp;
      v2f b1v; b1v.x = w1[k]; b1v.y = w1[k + 1];
      v2f b2v; b2v.x = w2[k]; b2v.y = w2[k + 1];
      a1[nt] = __builtin_amdgcn_wmma_f32_16x16x4_f32(false, av, false, b1v,
                                                     (short)0, a1[nt], false, false);
      a2[nt] = __builtin_amdgcn_wmma_f32_16x16x4_f32(false, av, false, b2v,
                                                     (short)0, a2[nt], false, false);
    }
  }
  const int b = row0 >> 16;
  const int i = (row0 >> 8) & 255;
  const int j0 = row0 & 255;
  const int n = lane & 15;
  const int mbase = (lane >> 4) << 3;
  const float mi = mask[(b << 8) + i];
  const float scale = 0.17677669529663687f; // 1/sqrt(DF=32)
#pragma unroll
  for (int nt = 0; nt < 4; ++nt) {
    const int c = col0 + nt * 16 + n;
    const float qv = Q[((b << 8) + i) * DXX + c];
    const float b1 = badd[c], b2 = bmul[c];
#pragma unroll
    for (int v = 0; v < 8; ++v) {
      const int j = j0 + mbase + v;
      const float kv = Km[((b << 8) + j) * DXX + c];
      const float em = mi * mask[(b << 8) + j];
      const float yqk = qv * kv * scale;
      const float e1 = (a1[nt][v] + b1) * em;
      const float e2 = (a2[nt][v] + b2) * em;
      Ybig[(size_t)(row0 + mbase + v) * DXX + c] = e1 + e2 * yqk + yqk;
    }
  }
}

// ------- e_out GEMM with y_e FiLM folded into A-load + Em mask + E residual -------
// one wave computes the full 16x64 output row-block: FiLM'd Ybig row read ONCE.
__global__ void k_eout(const float* __restrict__ Ybig, const float* __restrict__ ye1,
                       const float* __restrict__ ye2, const float* __restrict__ W,
                       const float* __restrict__ bias, const float* __restrict__ E,
                       const float* __restrict__ mask, float* __restrict__ er) {
  const int lane = threadIdx.x;
  const int row0 = blockIdx.y << 4;
  const int b = row0 >> 16;
  const int r = lane & 15;
  const int kp = (lane >> 4) << 1;
  const float* arow = Ybig + (size_t)(row0 + r) * DXX;
  const float* y1 = ye1 + (b << 8);
  const float* y2 = ye2 + (b << 8);
  v8f acc[4] = {};
  for (int k = 0; k < DXX; k += 4) {
    const int k0 = k + kp;
    v2f av;
    av.x = y1[k0]     + (y2[k0]     + 1.f) * arow[k0];
    av.y = y1[k0 + 1] + (y2[k0 + 1] + 1.f) * arow[k0 + 1];
#pragma unroll
    for (int nt = 0; nt < 4; ++nt) {
      const float* w = W + (size_t)(nt * 16 + r) * DXX + kp;
      v2f bv; bv.x = w[k]; bv.y = w[k + 1];
      acc[nt] = __builtin_amdgcn_wmma_f32_16x16x4_f32(false, av, false, bv,
                                                      (short)0, acc[nt], false, false);
    }
  }
  const int i = (row0 >> 8) & 255;
  const int j0 = row0 & 255;
  const int n = lane & 15;
  const int mbase = (lane >> 4) << 3;
  const float mi = mask[(b << 8) + i];
#pragma unroll
  for (int nt = 0; nt < 4; ++nt) {
    const int c = nt * 16 + n;
    const float bi = bias[c];
#pragma unroll
    for (int v = 0; v < 8; ++v) {
      const float em = mi * mask[(b << 8) + j0 + mbase + v];
      const size_t idx = (size_t)(row0 + mbase + v) * DEE + c;
      er[idx] = (acc[nt][v] + bi) * em + E[idx];
    }
  }
}

// ------- per-feature online softmax over keys + V accumulation + y_x FiLM -------
__global__ void k_attn(const float* __restrict__ Ybig, const float* __restrict__ V,
                       const float* __restrict__ mask, const float* __restrict__ yx1,
                       const float* __restrict__ yx2, float* __restrict__ xa) {
  const int blk = blockIdx.x;          // b*N + i
  const int b = blk >> 8;
  const int c = threadIdx.x;           // 0..255 feature
  const float* yrow = Ybig + (size_t)blk * (NN * DXX) + c;
  const float* vb = V + (size_t)(b << 8) * DXX + c;
  const float* mk = mask + (b << 8);
  float m = -3.4e38f, s = 0.f, w = 0.f;
  for (int j = 0; j < NN; ++j) {
    if (mk[j] > 0.f) {
      const float yv = yrow[(size_t)j * DXX];
      const float nm = fmaxf(m, yv);
      const float sc = __expf(m - nm);
      const float e  = __expf(yv - nm);
      s = s * sc + e;
      w = w * sc + e * vb[(size_t)j * DXX];
      m = nm;
    }
  }
  const float wv = (s > 0.f) ? (w / s) : 0.f;
  const float a1 = yx1[(b << 8) + c], a2 = yx2[(b << 8) + c];
  xa[(size_t)blk * DXX + c] = a1 + (a2 + 1.f) * wv;
}

// ---------------- LayerNorm: one block (D threads) per row ----------------
__global__ void k_layernorm(const float* __restrict__ x, const float* __restrict__ g,
                            const float* __restrict__ bta, float* __restrict__ out, int D) {
  const int row = blockIdx.x;
  const int t = threadIdx.x;
  __shared__ float sb[256];
  const float v = x[(size_t)row * D + t];
  sb[t] = v; __syncthreads();
  for (int s = blockDim.x >> 1; s > 0; s >>= 1) {
    if (t < s) sb[t] += sb[t + s];
    __syncthreads();
  }
  const float mu = sb[0] / (float)D;
  __syncthreads();
  const float d = v - mu;
  sb[t] = d * d; __syncthreads();
  for (int s = blockDim.x >> 1; s > 0; s >>= 1) {
    if (t < s) sb[t] += sb[t + s];
    __syncthreads();
  }
  const float rs = rsqrtf(sb[0] / (float)D + 1e-5f);
  out[(size_t)row * D + t] = d * rs * g[t] + bta[t];
}

// ---------------- small y-conditioning linears (4x256 each) ----------------
__global__ void k_ylin(const float* __restrict__ y,
                       const float* __restrict__ w_ea, const float* __restrict__ b_ea,
                       const float* __restrict__ w_em, const float* __restrict__ b_em,
                       const float* __restrict__ w_xa, const float* __restrict__ b_xa,
                       const float* __restrict__ w_xm, const float* __restrict__ b_xm,
                       float* ye1, float* ye2, float* yx1, float* yx2) {
  const int b = blockIdx.x, c = threadIdx.x;
  const float* yb = y + b * DYY;
  float s0 = b_ea[c], s1 = b_em[c], s2 = b_xa[c], s3 = b_xm[c];
  for (int k = 0; k < DYY; ++k) {
    const float yv = yb[k];
    s0 += yv * w_ea[c * DYY + k];
    s1 += yv * w_em[c * DYY + k];
    s2 += yv * w_xa[c * DYY + k];
    s3 += yv * w_xm[c * DYY + k];
  }
  ye1[b * 256 + c] = s0; ye2[b * 256 + c] = s1;
  yx1[b * 256 + c] = s2; yx2[b * 256 + c] = s3;
}

// ---------------- PNA reductions ----------------
__global__ void k_pnaX(const float* __restrict__ X, float* __restrict__ zX) {
  const int b = blockIdx.x, c = threadIdx.x; // 256 threads
  float s = 0.f, s2 = 0.f, lo = 3.4e38f, hi = -3.4e38f;
  for (int i = 0; i < NN; ++i) {
    const float v = X[((b << 8) + i) * DXX + c];
    s += v; s2 += v * v; lo = fminf(lo, v); hi = fmaxf(hi, v);
  }
  const float mean = s / (float)NN;
  const float var = (s2 - (float)NN * mean * mean) / (float)(NN - 1);
  zX[b * 1024 + c]       = mean;
  zX[b * 1024 + 256 + c] = lo;
  zX[b * 1024 + 512 + c] = hi;
  zX[b * 1024 + 768 + c] = sqrtf(fmaxf(var, 0.f));
}

__global__ void k_pnaE1(const float* __restrict__ E, float* __restrict__ pm,
                        float* __restrict__ ps, float* __restrict__ pmn,
                        float* __restrict__ pmx) {
  const int bi = blockIdx.x;           // b*N + i
  const int c = threadIdx.x;           // 0..63
  float s = 0.f, s2 = 0.f, lo = 3.4e38f, hi = -3.4e38f;
  for (int j = 0; j < NN; ++j) {
    const float v = E[((size_t)bi * NN + j) * DEE + c];
    s += v; s2 += v * v; lo = fminf(lo, v); hi = fmaxf(hi, v);
  }
  const float mean = s / (float)NN;
  const float var = (s2 - (float)NN * mean * mean) / (float)(NN - 1);
  pm[bi * DEE + c] = mean;
  ps[bi * DEE + c] = sqrtf(fmaxf(var, 0.f));
  pmn[bi * DEE + c] = lo;
  pmx[bi * DEE + c] = hi;
}

__global__ void k_pnaE2(const float* __restrict__ pm, const float* __restrict__ ps,
                        const float* __restrict__ pmn, const float* __restrict__ pmx,
                        float* __restrict__ zE) {
  const int b = blockIdx.x, c = threadIdx.x; // 64 threads
  float sm = 0.f, ss = 0.f, ss2 = 0.f, lo = 3.4e38f, hi = -3.4e38f;
  for (int i = 0; i < NN; ++i) {
    const int idx = ((b << 8) + i) * DEE + c;
    sm += pm[idx];
    const float sv = ps[idx];
    ss += sv; ss2 += sv * sv;
    lo = fminf(lo, pmn[idx]); hi = fmaxf(hi, pmx[idx]);
  }
  const float ms = ss / (float)NN;
  const float var = (ss2 - (float)NN * ms * ms) / (float)(NN - 1);
  zE[b * 256 + c]       = sm / (float)NN;
  zE[b * 256 + 64 + c]  = lo;
  zE[b * 256 + 128 + c] = hi;
  zE[b * 256 + 192 + c] = sqrtf(fmaxf(var, 0.f));
}

// ---------------- full y path (tiny): one block of 256 threads ----------------
__global__ void k_yfinal(const float* __restrict__ y, const float* __restrict__ zX,
                         const float* __restrict__ zE,
                         const float* yyw, const float* yyb,
                         const float* xyw, const float* xyb,
                         const float* eyw, const float* eyb,
                         const float* o1w, const float* o1b,
                         const float* o2w, const float* o2b,
                         const float* g1, const float* bb1,
                         const float* f1w, const float* f1b,
                         const float* f2w, const float* f2b,
                         const float* g2, const float* bb2,
                         float* __restrict__ out) {
  __shared__ float sA[256], sB[256], sY[256], sH[1024];
  __shared__ float sMu[4], sRs[4];
  const int t = threadIdx.x;
  const int b = t >> 6, c = t & 63;
  // new_y = y@y_y^T + pna(X) + pna(E) (biases included)
  float s = yyb[c] + xyb[c] + eyb[c];
  for (int k = 0; k < 64; ++k)   s += y[b * 64 + k]   * yyw[c * 64 + k];
  for (int k = 0; k < 1024; ++k) s += zX[b * 1024 + k] * xyw[c * 1024 + k];
  for (int k = 0; k < 256; ++k)  s += zE[b * 256 + k]  * eyw[c * 256 + k];
  sA[t] = s; __syncthreads();
  // y_out2(relu(y_out1(new_y)))
  float t1 = o1b[c];
  for (int k = 0; k < 64; ++k) t1 += sA[b * 64 + k] * o1w[c * 64 + k];
  t1 = fmaxf(t1, 0.f);
  sB[t] = t1; __syncthreads();
  float t2 = o2b[c];
  for (int k = 0; k < 64; ++k) t2 += sB[b * 64 + k] * o2w[c * 64 + k];
  const float r = y[b * 64 + c] + t2;
  __syncthreads();
  sA[t] = r; __syncthreads();
  if (t < 4) {
    float mu = 0.f; for (int k = 0; k < 64; ++k) mu += sA[t * 64 + k]; mu /= 64.f;
    float var = 0.f; for (int k = 0; k < 64; ++k) { const float d = sA[t * 64 + k] - mu; var += d * d; }
    sMu[t] = mu; sRs[t] = rsqrtf(var / 64.f + 1e-5f);
  }
  __syncthreads();
  const float yln = (r - sMu[b]) * sRs[b] * g1[c] + bb1[c];
  sY[t] = yln; __syncthreads();
  // FFy hidden (4 x 256): thread t = hidden channel
  for (int bb = 0; bb < 4; ++bb) {
    float h = f1b[t];
    for (int k = 0; k < 64; ++k) h += sY[bb * 64 + k] * f1w[t * 64 + k];
    sH[bb * 256 + t] = fmaxf(h, 0.f);
  }
  __syncthreads();
  float o = f2b[c];
  for (int k = 0; k < 256; ++k) o += sH[b * 256 + k] * f2w[c * 256 + k];
  const float r2 = sY[t] + o;
  __syncthreads();
  sA[t] = r2; __syncthreads();
  if (t < 4) {
    float mu = 0.f; for (int k = 0; k < 64; ++k) mu += sA[t * 64 + k]; mu /= 64.f;
    float var = 0.f; for (int k = 0; k < 64; ++k) { const float d = sA[t * 64 + k] - mu; var += d * d; }
    sMu[t] = mu; sRs[t] = rsqrtf(var / 64.f + 1e-5f);
  }
  __syncthreads();
  out[t] = (r2 - sMu[b]) * sRs[b] * g2[c] + bb2[c];
}

// ============================================================================
extern "C" void kernel_launch(void* const* d_in, const int* in_sizes, int n_in,
                              void* d_out, int out_size, void* d_ws, size_t ws_size,
                              hipStream_t stream) {
  (void)in_sizes; (void)n_in; (void)out_size; (void)ws_size;
  const float* X    = (const float*)d_in[IN_X];
  const float* E    = (const float*)d_in[IN_E];
  const float* y    = (const float*)d_in[IN_Y];
  const float* mask = (const float*)d_in[IN_MASK];
#define P(i) ((const float*)d_in[(i)])
  float* ws = (float*)d_ws;
  float* out = (float*)d_out;

  float* Q   = ws + OFF_Q;
  float* Kb  = ws + OFF_K;
  float* V   = ws + OFF_V;
  float* xa  = ws + OFF_XA;
  float* xr  = ws + OFF_XR;
  float* xln = ws + OFF_XLN;
  float* xh  = ws + OFF_XH;
  float* xr2 = ws + OFF_XR2;
  float* ye1 = ws + OFF_YE1;
  float* ye2 = ws + OFF_YE2;
  float* yx1 = ws + OFF_YX1;
  float* yx2 = ws + OFF_YX2;
  float* zX  = ws + OFF_ZX;
  float* pm  = ws + OFF_PM;
  float* ps  = ws + OFF_PS;
  float* pmn = ws + OFF_PMN;
  float* pmx = ws + OFF_PMX;
  float* zE  = ws + OFF_ZE;
  float* er  = ws + OFF_ER;     // also er2 (aliased after eln computed)
  float* eln = ws + OFF_ELN;
  float* Ybig = ws + OFF_YBIG;  // also ffE hidden (aliased after Ybig consumed)

  const int MX = BSZ * NN;            // 1024 X-rows
  const int ME = BSZ * NN * NN;       // 262144 E-rows

  // y conditioning linears
  k_ylin<<<BSZ, 256, 0, stream>>>(y, P(P_y_e_add_w), P(P_y_e_add_b),
                                  P(P_y_e_mul_w), P(P_y_e_mul_b),
                                  P(P_y_x_add_w), P(P_y_x_add_b),
                                  P(P_y_x_mul_w), P(P_y_x_mul_b),
                                  ye1, ye2, yx1, yx2);

  // Q/K/V projections (masked)
  k_gemm<<<dim3(DXX / 64, MX / 16), 32, 0, stream>>>(X, P(P_q_w), P(P_q_b), nullptr, mask, Q,  DXX, DXX, 0);
  k_gemm<<<dim3(DXX / 64, MX / 16), 32, 0, stream>>>(X, P(P_k_w), P(P_k_b), nullptr, mask, Kb, DXX, DXX, 0);
  k_gemm<<<dim3(DXX / 64, MX / 16), 32, 0, stream>>>(X, P(P_v_w), P(P_v_b), nullptr, mask, V,  DXX, DXX, 0);

  // fused e_add/e_mul GEMMs + QK FiLM -> Ybig
  k_e12<<<dim3(DXX / 64, ME / 16), 32, 0, stream>>>(E, P(P_e_add_w), P(P_e_add_b),
                                                    P(P_e_mul_w), P(P_e_mul_b),
                                                    Q, Kb, mask, Ybig);

  // per-feature softmax over keys + V + y_x FiLM -> xa
  k_attn<<<MX, 256, 0, stream>>>(Ybig, V, mask, yx1, yx2, xa);

  // x_out + mask + X residual -> xr ; LN -> xln ; FFX ; LN -> Xo
  k_gemm<<<dim3(DXX / 64, MX / 16), 32, 0, stream>>>(xa, P(P_x_out_w), P(P_x_out_b), X, mask, xr, DXX, DXX, 0);
  k_layernorm<<<MX, DXX, 0, stream>>>(xr, P(P_lnX1_g), P(P_lnX1_b), xln, DXX);
  k_gemm<<<dim3(1024 / 64, MX / 16), 32, 0, stream>>>(xln, P(P_ffX1_w), P(P_ffX1_b), nullptr, nullptr, xh, 1024, DXX, 1);
  k_gemm<<<dim3(DXX / 64, MX / 16), 32, 0, stream>>>(xh, P(P_ffX2_w), P(P_ffX2_b), xln, nullptr, xr2, DXX, 1024, 0);
  k_layernorm<<<MX, DXX, 0, stream>>>(xr2, P(P_lnX2_g), P(P_lnX2_b), out + OUT_XO, DXX);

  // e_out with y_e FiLM folded in + Em mask + E residual -> er ; LN -> eln
  k_eout<<<dim3(1, ME / 16), 32, 0, stream>>>(Ybig, ye1, ye2, P(P_e_out_w), P(P_e_out_b), E, mask, er);
  k_layernorm<<<ME, DEE, 0, stream>>>(er, P(P_lnE1_g), P(P_lnE1_b), eln, DEE);

  // FFE (hidden aliases Ybig; Ybig fully consumed above) ; LN -> Eo
  k_gemm<<<dim3(256 / 64, ME / 16), 32, 0, stream>>>(eln, P(P_ffE1_w), P(P_ffE1_b), nullptr, nullptr, Ybig, 256, DEE, 1);
  k_gemm<<<dim3(1, ME / 16), 32, 0, stream>>>(Ybig, P(P_ffE2_w), P(P_ffE2_b), eln, nullptr, er, DEE, 256, 0);
  k_layernorm<<<ME, DEE, 0, stream>>>(er, P(P_lnE2_g), P(P_lnE2_b), out + OUT_EO, DEE);

  // PNA reductions + full y path
  k_pnaX<<<BSZ, 256, 0, stream>>>(X, zX);
  k_pnaE1<<<BSZ * NN, DEE, 0, stream>>>(E, pm, ps, pmn, pmx);
  k_pnaE2<<<BSZ, DEE, 0, stream>>>(pm, ps, pmn, pmx, zE);
  k_yfinal<<<1, 256, 0, stream>>>(y, zX, zE,
                                  P(P_y_y_w), P(P_y_y_b),
                                  P(P_x_y_w), P(P_x_y_b),
                                  P(P_e_y_w), P(P_e_y_b),
                                  P(P_y_out1_w), P(P_y_out1_b),
                                  P(P_y_out2_w), P(P_y_out2_b),
                                  P(P_lny1_g), P(P_lny1_b),
                                  P(P_ffy1_w), P(P_ffy1_b),
                                  P(P_ffy2_w), P(P_ffy2_b),
                                  P(P_lny2_g), P(P_lny2_b),
                                  out + OUT_YO);
#undef P
}